// LightGCNConv_32701880991783
// MI455X (gfx1250) — compile-verified
//
#include <hip/hip_runtime.h>

// -------------------------------------------------------------------------
// COO SpMM: out[row[e]] += val[e] * x[col[e]]   (N=100000, E=3200000, D=64)
//
// Roofline: 0.41 GFLOP vs ~64-90 MB HBM traffic -> gather/scatter bound.
// x (25.6 MB) and out (25.6 MB) are L2-resident (192 MB L2); atomics resolve
// at the L2 atomic units. VALU budget (~300M lane-ops for a ~4us kernel)
// means the inner loop must stay lean: per 4 output elements we spend
// 1 ds_load_b128 (metadata) + 1 global_load_b128 (gather) + 4 v_mul +
// 4 global_atomic_add_f32 + addressing.
//
// Layout: 16 lanes per edge, one float4 per lane -> fully coalesced 256B
// accesses for both the x gather and the out atomic scatter. Edge metadata
// is staged per-block into LDS via the async-to-LDS path (ASYNCcnt).
// -------------------------------------------------------------------------

#define EPB 256  // edges staged per block (256 threads, 16 threads/edge)

#if defined(__HIP_DEVICE_COMPILE__) && __has_builtin(__builtin_amdgcn_global_load_async_to_lds_b32)
#define HAVE_ASYNC_LDS 1
#else
#define HAVE_ASYNC_LDS 0
#endif

__device__ __forceinline__ void wait_asynccnt0() {
#if defined(__HIP_DEVICE_COMPILE__)
#if __has_builtin(__builtin_amdgcn_s_wait_asynccnt)
  __builtin_amdgcn_s_wait_asynccnt(0);
#else
  asm volatile("s_wait_asynccnt 0" ::: "memory");
#endif
#endif
}

// Fire-and-forget f32 atomic add -> global_atomic_add_f32 (no-return form,
// STOREcnt tracked, RMW at the L2 atomic units). No "memory" clobber: the
// atomics have no intra-kernel ordering requirement (out is never read), so
// the compiler may batch subsequent gather loads above them for ILP.
__device__ __forceinline__ void atomic_add_f32(float* p, float v) {
  asm volatile("global_atomic_add_f32 %0, %1, off" ::"v"(p), "v"(v));
}

__global__ void zero_f4(float4* __restrict__ o, int n4) {
  int i = blockIdx.x * blockDim.x + threadIdx.x;
  if (i < n4) o[i] = make_float4(0.f, 0.f, 0.f, 0.f);
}

__launch_bounds__(256)
__global__ void spmm_coo(const float* __restrict__ x,
                         const int* __restrict__ row,
                         const int* __restrict__ col,
                         const float* __restrict__ val,
                         float* __restrict__ out,
                         int nE) {
  // Packed per-edge metadata: {row, col, val, pad} -> one ds_load_b128 each.
  __shared__ int4 s_meta[EPB];

  const int t    = threadIdx.x;
  const int base = blockIdx.x * EPB;

  // ---- Stage this block's edge metadata into LDS (ASYNCcnt path) ----
  {
    const int e = base + t;
    if (e < nE) {
#if HAVE_ASYNC_LDS
      __builtin_amdgcn_global_load_async_to_lds_b32(
          (int*)(row + e), (int*)&s_meta[t].x, 0, 0);
      __builtin_amdgcn_global_load_async_to_lds_b32(
          (int*)(col + e), (int*)&s_meta[t].y, 0, 0);
      __builtin_amdgcn_global_load_async_to_lds_b32(
          (int*)(val + e), (int*)&s_meta[t].z, 0, 0);
#else
      s_meta[t].x = row[e];
      s_meta[t].y = col[e];
      s_meta[t].z = __float_as_int(val[e]);
#endif
    }
#if HAVE_ASYNC_LDS
    wait_asynccnt0();
#endif
  }
  __syncthreads();

  // ---- 16 edges per thread, one float4 column-chunk each ----
  const int d4   = t & 15;  // which float4 (0..15) of the 64-float row
  const int egrp = t >> 4;  // edge sub-group within the block (0..15)
  const float4* __restrict__ x4b = (const float4*)x + d4;  // + c*16 per edge
  float* __restrict__ ob = out + (d4 << 2);                // + r*64 per edge

  if (base + EPB <= nE) {
    // Fast path: full tile, no per-iteration guards.
#pragma unroll 4
    for (int k = 0; k < EPB / 16; ++k) {
      const int  el = (k << 4) + egrp;  // broadcast within each 16-lane group
      const int4 m  = s_meta[el];       // ds_load_b128: {r, c, val_bits, -}
      const float  v  = __int_as_float(m.z);
      const float4 xv = x4b[m.y * 16];  // coalesced 256B gather per group
      float* o = ob + ((size_t)m.x << 6);
      atomic_add_f32(o + 0, v * xv.x);
      atomic_add_f32(o + 1, v * xv.y);
      atomic_add_f32(o + 2, v * xv.z);
      atomic_add_f32(o + 3, v * xv.w);
    }
  } else {
    // Tail tile (not hit for E=3.2M, kept for generality).
    for (int k = 0; k < EPB / 16; ++k) {
      const int el = (k << 4) + egrp;
      if (base + el < nE) {
        const int4 m  = s_meta[el];
        const float  v  = __int_as_float(m.z);
        const float4 xv = x4b[m.y * 16];
        float* o = ob + ((size_t)m.x << 6);
        atomic_add_f32(o + 0, v * xv.x);
        atomic_add_f32(o + 1, v * xv.y);
        atomic_add_f32(o + 2, v * xv.z);
        atomic_add_f32(o + 3, v * xv.w);
      }
    }
  }
}

extern "C" void kernel_launch(void* const* d_in, const int* in_sizes, int n_in,
                              void* d_out, int out_size, void* d_ws, size_t ws_size,
                              hipStream_t stream) {
  (void)n_in; (void)d_ws; (void)ws_size;
  const float* x   = (const float*)d_in[0];
  const int*   row = (const int*)d_in[1];
  const int*   col = (const int*)d_in[2];
  const float* val = (const float*)d_in[3];
  float*       out = (float*)d_out;

  const int nE = in_sizes[1];

  // Atomic accumulation requires a zeroed output every call (d_out is poisoned).
  const int n4 = out_size / 4;
  zero_f4<<<(n4 + 255) / 256, 256, 0, stream>>>((float4*)out, n4);

  const int blocks = (nE + EPB - 1) / EPB;
  spmm_coo<<<blocks, 256, 0, stream>>>(x, row, col, val, out, nE);
}